// MultiHeadAttentionQuantum_65481071403167
// MI455X (gfx1250) — compile-verified
//
#include <hip/hip_runtime.h>
#include <math.h>

// Problem dims (fixed by the reference)
#define BATCH 8
#define SEQ   4096
#define EMB   256
#define NH    64
#define DK    4

typedef __attribute__((ext_vector_type(2))) float v2f;
typedef __attribute__((ext_vector_type(8))) float v8f;
typedef __attribute__((ext_vector_type(4))) unsigned int u32x4;
typedef __attribute__((ext_vector_type(8))) int i32x8;
typedef __attribute__((ext_vector_type(4))) int i32x4;

#if defined(__has_builtin)
#if __has_builtin(__builtin_amdgcn_tensor_load_to_lds) && \
    __has_builtin(__builtin_amdgcn_s_wait_tensorcnt)
#define USE_TDM 1
#endif
#if __has_builtin(__builtin_amdgcn_sched_barrier)
// Allow ALU/VALU/SALU/VMEM/trans to cross; pin DS (0x380) and WMMA (0x8).
#define PIPELINE_FENCE() __builtin_amdgcn_sched_barrier(0x477)
#endif
#endif
#ifndef USE_TDM
#define USE_TDM 0
#endif
#ifndef PIPELINE_FENCE
#define PIPELINE_FENCE()
#endif

// ---------------------------------------------------------------------------
// Kernel A: quantum heads + per-token attention over the HEAD axis.
// One wave (32 lanes) per token; each lane owns heads (2*lane, 2*lane+1).
// Writes result in the permuted [B, H, S, DK] layout directly into `y`
// (which row-major-flattens to the logical [B*S, E] matrix fed to the GEMM).
// ---------------------------------------------------------------------------
__global__ __launch_bounds__(256) void qattn_kernel(const float* __restrict__ x,
                                                    const float* __restrict__ theta,
                                                    float* __restrict__ y) {
  __shared__ __align__(16) float q_lds[8][NH][DK];   // 8 KB: per-wave Q staging
  const int lane = threadIdx.x & 31;
  const int wave = threadIdx.x >> 5;
  const int tok  = blockIdx.x * 8 + wave;            // global token = b*SEQ + s
  const int b    = tok >> 12;                        // / 4096 (same for whole block)

  const float t0 = theta[0], t1 = theta[1], t2 = theta[2], t3 = theta[3];

  // x[tok, 8*lane .. 8*lane+7] : head h0 = 2*lane uses dims 0..3, h1 dims 4..7
  const float4* xp = (const float4*)(x + (size_t)tok * EMB + lane * 8);
  float4 xa = xp[0];
  float4 xb = xp[1];

  // q = cumprod(cos(x + theta)) along d  (theta broadcast over heads)
  float q0[4], q1[4];
  q0[0] = __cosf(xa.x + t0);
  q0[1] = q0[0] * __cosf(xa.y + t1);
  q0[2] = q0[1] * __cosf(xa.z + t2);
  q0[3] = q0[2] * __cosf(xa.w + t3);
  q1[0] = __cosf(xb.x + t0);
  q1[1] = q1[0] * __cosf(xb.y + t1);
  q1[2] = q1[1] * __cosf(xb.z + t2);
  q1[3] = q1[2] * __cosf(xb.w + t3);

  float4* qs = (float4*)&q_lds[wave][lane * 2][0];
  qs[0] = make_float4(q0[0], q0[1], q0[2], q0[3]);
  qs[1] = make_float4(q1[0], q1[1], q1[2], q1[3]);
  __syncthreads();

  // Flash-style online softmax over the 64 "keys" (heads of the same token).
  float m0 = -INFINITY, l0 = 0.f, m1 = -INFINITY, l1 = 0.f;
  float o0[4] = {0.f, 0.f, 0.f, 0.f};
  float o1[4] = {0.f, 0.f, 0.f, 0.f};
#pragma unroll 4
  for (int t = 0; t < NH; ++t) {
    float4 qt = *(const float4*)&q_lds[wave][t][0];  // LDS broadcast read
    float s0 = 0.5f * (q0[0] * qt.x + q0[1] * qt.y + q0[2] * qt.z + q0[3] * qt.w);
    float s1 = 0.5f * (q1[0] * qt.x + q1[1] * qt.y + q1[2] * qt.z + q1[3] * qt.w);

    float nm0 = fmaxf(m0, s0);
    float c0  = __expf(m0 - nm0);
    float p0  = __expf(s0 - nm0);
    l0 = l0 * c0 + p0;
    o0[0] = o0[0] * c0 + p0 * qt.x;
    o0[1] = o0[1] * c0 + p0 * qt.y;
    o0[2] = o0[2] * c0 + p0 * qt.z;
    o0[3] = o0[3] * c0 + p0 * qt.w;
    m0 = nm0;

    float nm1 = fmaxf(m1, s1);
    float c1  = __expf(m1 - nm1);
    float p1  = __expf(s1 - nm1);
    l1 = l1 * c1 + p1;
    o1[0] = o1[0] * c1 + p1 * qt.x;
    o1[1] = o1[1] * c1 + p1 * qt.y;
    o1[2] = o1[2] * c1 + p1 * qt.z;
    o1[3] = o1[3] * c1 + p1 * qt.w;
    m1 = nm1;
  }
  float i0 = 1.f / l0, i1 = 1.f / l1;

  // Overwrite this wave's staging slice with its attention output.
  // Safe: DS ops from the same wave execute in order (loads above complete first).
  qs[0] = make_float4(o0[0] * i0, o0[1] * i0, o0[2] * i0, o0[3] * i0);
  qs[1] = make_float4(o1[0] * i1, o1[1] * i1, o1[2] * i1, o1[3] * i1);
  __syncthreads();

  // Cooperative permuted store: for a fixed head, the block's 8 consecutive
  // tokens are contiguous in [B,H,S,DK] -> 128B runs (8 lanes' float4 each).
  const int sBase = (blockIdx.x * 8) & (SEQ - 1);
#pragma unroll
  for (int r = 0; r < 2; ++r) {
    int idx = threadIdx.x + (r << 8);  // 0..511
    int h   = idx >> 3;
    int si  = idx & 7;
    float4 v = *(const float4*)&q_lds[si][h][0];
    *(float4*)(y + (((size_t)(b * NH + h)) * SEQ + sBase + si) * DK) = v;
  }
}

// ---------------------------------------------------------------------------
// Kernel B: in-place GEMM  y[r, :] = y[r, :] @ W^T  via V_WMMA_F32_16X16X4_F32.
// Block = 8 waves: 128 M-rows x 256 N-cols, K tiled by 64 through LDS.
// W^T chunks are fetched by the Tensor Data Mover (2D tile, DMA-applied LDS
// padding -> row stride 68 dwords, bank-conflict-free B-fragment reads).
// Inner loop is software-pipelined: B fragments for k-step kk+1 are loaded
// (double-buffered) while the 16 WMMAs of k-step kk execute; sched_barrier
// fences pin DS and WMMA so the scheduler cannot re-serialize the loads.
// In-place is safe: a block writes only the rows it (alone) reads.
// ---------------------------------------------------------------------------
#define LDS_ROW 68  // 64 K-values + 4 dwords pad

__global__ __launch_bounds__(256) void gemm_wt_kernel(float* __restrict__ y,
                                                      const float* __restrict__ W) {
  __shared__ __align__(16) float bLds[EMB * LDS_ROW];  // 68 KB
  const int lane = threadIdx.x & 31;
  const int ln   = lane & 15;                          // M / N index within tile
  const int hi   = lane >> 4;                          // upper half-wave flag
  const int wave = threadIdx.x >> 5;
  const size_t rowBase = (size_t)blockIdx.x * 128 + wave * 16;

  v8f acc[16];
#pragma unroll
  for (int i = 0; i < 16; ++i)
#pragma unroll
    for (int j = 0; j < 8; ++j) acc[i][j] = 0.f;

  for (int k0 = 0; k0 < EMB; k0 += 64) {
#if USE_TDM
    if (threadIdx.x < 32) {  // one wave issues the DMA (TDM ignores EXEC)
      // ---- Tensor DMA descriptor (D#): 2D tile 64(K) x 256(N) of W^T ----
      unsigned lds_base = (unsigned)(size_t)&bLds[0];  // LDS aperture: low 32 bits
      unsigned long long ga = (unsigned long long)(size_t)(W + k0);
      u32x4 g0;
      g0[0] = 1u;                                  // count=1 (valid), no gather
      g0[1] = lds_base;                            // lds_addr
      g0[2] = (unsigned)(ga & 0xffffffffu);        // global_addr[31:0]
      g0[3] = (unsigned)((ga >> 32) & 0x01ffffffu) // global_addr[56:32]
              | (2u << 30);                        // type=2 (image)
      i32x8 g1;
      g1[0] = (2 << 16)     // data_size = 4 bytes
            | (1 << 20)     // pad_enable
            | (5 << 22)     // pad_interval: 64 dwords
            | (3 << 25);    // pad_amount:   4 dwords  -> LDS row stride 68 dw
      g1[1] = (int)(256u << 16);  // tensor_dim0 = 256 (low 16 @ bits 63:48)
      g1[2] = (int)(256u << 16);  // dim0 hi=0 | tensor_dim1 = 256 (bits 95:80)
      g1[3] = (int)(64u << 16);   // dim1 hi=0 | tile_dim0 = 64 (bits 127:112)
      g1[4] = 256;                // tile_dim1 = 256 | tile_dim2 = 0
      g1[5] = 256;                // tensor_dim0_stride[31:0] = 256 elements
      g1[6] = 0;                  // stride0 hi | tensor_dim1_stride lo
      g1[7] = 0;
      i32x4 z4 = {0, 0, 0, 0};                  // groups 2/3 unused (2D tensor)
      i32x8 z8 = {0, 0, 0, 0, 0, 0, 0, 0};      // group 4 unused (6-arg form)
      __builtin_amdgcn_tensor_load_to_lds(g0, g1, z4, z4, z8, 0);
      __builtin_amdgcn_s_wait_tensorcnt(0);
    }
    __syncthreads();
#else
    {  // cooperative fallback: thread n stages W[n][k0..k0+63] (coalesced)
      const float4* wrow = (const float4*)(W + (size_t)threadIdx.x * EMB + k0);
      float* dst = &bLds[threadIdx.x * LDS_ROW];
#pragma unroll
      for (int j4 = 0; j4 < 16; ++j4) {
        float4 wv = wrow[j4];
        *(float4*)(dst + j4 * 4) = wv;  // 272B row stride keeps 16B alignment
      }
    }
    __syncthreads();
#endif

    // -------- software-pipelined WMMA sweep over the 16 k-steps --------
    v2f bb[2][16];  // double-buffered B fragments (fully unrolled -> VGPRs)
#pragma unroll
    for (int nt = 0; nt < 16; ++nt)
      bb[0][nt] = *(const v2f*)&bLds[(nt * 16 + ln) * LDS_ROW + 2 * hi];

#pragma unroll
    for (int kk = 0; kk < 16; ++kk) {
      const int cur = kk & 1;
      if (kk < 15) {  // prefetch B fragments for k-step kk+1 into other buffer
#pragma unroll
        for (int nt = 0; nt < 16; ++nt)
          bb[cur ^ 1][nt] =
              *(const v2f*)&bLds[(nt * 16 + ln) * LDS_ROW + (kk + 1) * 4 + 2 * hi];
      }
      PIPELINE_FENCE();  // keep prefetch DS loads above this k-step's WMMAs

      // A fragment (16x4 f32): lane l, vgpr v -> A[l&15][2*hi + v]
      v2f a = *(const v2f*)(y + (rowBase + ln) * EMB + k0 + kk * 4 + 2 * hi);
#pragma unroll
      for (int nt = 0; nt < 16; ++nt)
        acc[nt] = __builtin_amdgcn_wmma_f32_16x16x4_f32(
            /*neg_a=*/false, a, /*neg_b=*/false, bb[cur][nt],
            /*c_mod=*/(short)0, acc[nt], /*reuse_a=*/false, /*reuse_b=*/false);
      PIPELINE_FENCE();  // keep this k-step's WMMAs from sinking past next loads
    }
    __syncthreads();
  }

  // Epilogue: D vgpr j, lane l -> C[j + 8*hi][l&15] per 16x16 tile.
#pragma unroll
  for (int nt = 0; nt < 16; ++nt)
#pragma unroll
    for (int j = 0; j < 8; ++j)
      y[(rowBase + j + 8 * hi) * EMB + nt * 16 + ln] = acc[nt][j];
}

// ---------------------------------------------------------------------------
extern "C" void kernel_launch(void* const* d_in, const int* in_sizes, int n_in,
                              void* d_out, int out_size, void* d_ws, size_t ws_size,
                              hipStream_t stream) {
  (void)in_sizes; (void)n_in; (void)out_size; (void)d_ws; (void)ws_size;
  const float* x     = (const float*)d_in[0];
  const float* theta = (const float*)d_in[1];
  const float* W     = (const float*)d_in[2];
  float* out = (float*)d_out;

  // Stage 1: attention -> permuted y written directly into d_out.
  qattn_kernel<<<(BATCH * SEQ) / 8, 256, 0, stream>>>(x, theta, out);
  // Stage 2: in-place  out = out @ W^T  (WMMA f32, TDM-staged W^T chunks).
  gemm_wt_kernel<<<(BATCH * SEQ) / 128, 256, 0, stream>>>(out, W);
}